// PixelVectorExtractor_11579231830593
// MI455X (gfx1250) — compile-verified
//
#include <hip/hip_runtime.h>

// PixelVectorExtractor: x (32,16,128,128) f32 -> out (32, 16384, 17, 9) f32
// out[n][h*128+w][c][hl*3+wl] = x[n,c,h+hl-1,w+wl-1] (zero-padded), c==16 -> 1.0
//
// Pure data movement: ~321 MB NT stores + 96 MB staged reads -> ~15us @ 23.3 TB/s.
// CDNA5 path: global_load_async_to_lds_b128 staging (ASYNCcnt) + s_wait_asynccnt,
// branch-free LUT-driven emit loop, coalesced non-temporal b128 stores.

#define N_   32
#define C_   16
#define H_   128
#define W_   128
#define ROWS 3
#define WP   136                       // LDS row pitch (floats); data at col 4..131
#define LDSF (C_ * ROWS * WP)          // 6528 floats = 26112 B
#define ONES LDSF                      // 128 floats of 1.0 right after the tile
#define OUTROW (W_ * (C_ + 1) * 9)     // 19584 floats per (n,h)

typedef float v4f __attribute__((ext_vector_type(4)));

__global__ __launch_bounds__(256)
void pixel_extract_kernel(const float* __restrict__ x, float* __restrict__ out) {
    __shared__ float buf[LDSF + 128];  // staged tile + ones region
    __shared__ int   lut[153];         // rm -> LDS base (add w to get element)

    const int tid = threadIdx.x;
    const int nh  = blockIdx.x;        // n*128 + h
    const int n   = nh >> 7;
    const int h   = nh & 127;

    // --- 1) zero the tile (zero padding + OOB rows), ones region, LUT ---
    for (int i = tid; i < LDSF; i += 256) buf[i] = 0.0f;
    if (tid < 128) buf[ONES + tid] = 1.0f;
    if (tid < 153) {
        int c  = tid / 9;
        int k  = tid - c * 9;
        int hl = k / 3;
        int wl = k - hl * 3;
        // buf[base + w] == x[n,c,h+hl-1,w+wl-1] (or 1.0 for the ones channel)
        lut[tid] = (c == C_) ? ONES : (c * ROWS + hl) * WP + wl + 3;
    }
    __syncthreads();

    // --- 2) async-stage x[n, 0..15, h-1..h+1, 0..127] into buf[c][r][4..131] ---
    // 1536 float4 chunks: q -> c = q/96, r = (q%96)/32, x4 = q%32.
    // Each wave32 covers one (c,r) row chunk => wave-uniform OOB predicate.
    #pragma unroll
    for (int i = 0; i < 6; ++i) {
        int q  = i * 256 + tid;
        int c  = q / 96;
        int rm = q - c * 96;
        int r  = rm >> 5;
        int x4 = rm & 31;
        int gr = h - 1 + r;
        if (0 <= gr && gr < H_) {
            const float* gp = x + ((((size_t)n * C_ + c) * H_ + gr) * W_ + (size_t)x4 * 4);
            // LDS byte offset ((c*3+r)*136 + 4 + 4*x4)*4 -> 16B aligned
            unsigned lp = (unsigned)(size_t)(&buf[(c * ROWS + r) * WP + 4 + x4 * 4]);
            asm volatile("global_load_async_to_lds_b128 %0, %1, off"
                         :: "v"(lp), "v"(gp) : "memory");
        }
    }
    asm volatile("s_wait_asynccnt 0x0" ::: "memory");
    __syncthreads();

    // --- 3) emit output row: branch-free, division-free, coalesced NT stores ---
    // t = e + j decomposes as t = w*153 + rm; carried incrementally.
    float* orow = out + (size_t)nh * OUTROW;
    int e  = tid * 4;                  // <= 1020
    int w0 = e / 153;                  // one divide total (outside the loop)
    int rm = e - w0 * 153;
    for (; e < OUTROW; e += 1024) {
        v4f v;
        #pragma unroll
        for (int j = 0; j < 4; ++j) {
            int rmj = rm + j;          // <= 155: at most one wrap
            int wj  = w0;
            if (rmj >= 153) { rmj -= 153; wj += 1; }
            v[j] = buf[lut[rmj] + wj];
        }
        __builtin_nontemporal_store(v, (v4f*)(orow + e));
        rm += 106;                     // 1024 - 6*153
        w0 += 6;
        if (rm >= 153) { rm -= 153; w0 += 1; }
    }
}

extern "C" void kernel_launch(void* const* d_in, const int* in_sizes, int n_in,
                              void* d_out, int out_size, void* d_ws, size_t ws_size,
                              hipStream_t stream) {
    const float* x = (const float*)d_in[0];
    float* out = (float*)d_out;
    (void)in_sizes; (void)n_in; (void)out_size; (void)d_ws; (void)ws_size;
    pixel_extract_kernel<<<N_ * H_, 256, 0, stream>>>(x, out);
}